// FMRIToROI_64647847739592
// MI455X (gfx1250) — compile-verified
//
#include <hip/hip_runtime.h>

// ---------------------------------------------------------------------------
// FMRIToROI forward, MI455X (gfx1250, wave32).
//
// Roofline: fc_w is [271753,1536] fp32 = 1.67 GB. Everything downstream of
// the big einsum is linear in t and v, so the 160-GFLOP GEMM collapses to
//   x_node[b,r] = xbar[b,:] . (segmean_r fc_w) + segmean_r fc_b
// i.e. one memory-bound streaming pass over fc_w (~72 us at 23.3 TB/s).
// The only real dense GEMM left (distance MLP, 13456x64x64) is done with
// V_WMMA_F32_16X16X4_F32 (fp32, exact), one wave per 16-row tile.
// ---------------------------------------------------------------------------

#define NRROI 116
#define VOXN  271753
#define CIN   1536            // C*D*D*D = 192*8
#define BB    8               // batch
#define TT    24              // time
#define NPAIR (NRROI * NRROI) // 13456 = 841*16

typedef __attribute__((ext_vector_type(2))) float v2f;
typedef __attribute__((ext_vector_type(8))) float v8f;

// ---------------- workspace offsets (in floats) ----------------
#define OFF_ROI_SUM 0                         // 116*1536 = 178176
#define OFF_CNT     178176                    // 116
#define OFF_BSUM    178292                    // 116
#define ZERO_FLOATS 178408                    // memset region end
#define OFF_XBAR    178408                    // 8*1536
#define OFF_POOLED  190696                    // 8*192
#define OFF_XNODE   192232                    // 8*116
#define OFF_FX      193160                    // 8*116
#define OFF_MD      194088                    // 13456
#define OFF_MMAT    207544                    // 64
#define OFF_CAT     207608                    // 8*308
#define OFF_HID     210072                    // 8*768

// ---------------------------------------------------------------------------
// 1) ROI voxel counts + segment-sum of fc_b (small, atomics from zeroed ws)
// ---------------------------------------------------------------------------
__global__ __launch_bounds__(256)
void k_count(const int* __restrict__ labels, const float* __restrict__ fc_b,
             float* __restrict__ cnt, float* __restrict__ bsum) {
  __shared__ float sc[NRROI];
  __shared__ float sb[NRROI];
  const int tid = threadIdx.x;
  for (int i = tid; i < NRROI; i += 256) { sc[i] = 0.f; sb[i] = 0.f; }
  __syncthreads();
  for (int v = blockIdx.x * 256 + tid; v < VOXN; v += gridDim.x * 256) {
    int lab = labels[v];
    atomicAdd(&sc[lab], 1.0f);
    atomicAdd(&sb[lab], fc_b[v]);
  }
  __syncthreads();
  for (int i = tid; i < NRROI; i += 256) {
    atomicAdd(&cnt[i],  sc[i]);
    atomicAdd(&bsum[i], sb[i]);
  }
}

// ---------------------------------------------------------------------------
// 2) THE dominant kernel: stream fc_w [VOXN,1536] once, segment-sum rows by
//    ROI label into roi_sum[116,1536]. Each block owns a 128-column tile and
//    a 4096-row chunk; per-thread-exclusive LDS accumulator bank [116][128]
//    (59 KB), then one global atomicAdd flush per (roi,col).
//    12 col-tiles x 67 chunks = 804 blocks of 4 waves -> saturates HBM.
// ---------------------------------------------------------------------------
#define RED_COLS 128
#define RED_ROWS 4096
__global__ __launch_bounds__(RED_COLS)
void k_reduce_fcw(const float* __restrict__ fc_w, const int* __restrict__ labels,
                  float* __restrict__ roi_sum) {
  __shared__ float acc[NRROI * RED_COLS];  // 59392 B
  const int tid = threadIdx.x;
  const int colbase = blockIdx.x * RED_COLS;
  const int row0 = blockIdx.y * RED_ROWS;
  int row1 = row0 + RED_ROWS; if (row1 > VOXN) row1 = VOXN;

  for (int i = tid; i < NRROI * RED_COLS; i += RED_COLS) acc[i] = 0.f;
  __syncthreads();

  const float* __restrict__ p = fc_w + (long)row0 * CIN + colbase + tid;
#pragma unroll 4
  for (int v = row0; v < row1; ++v) {
    int lab = labels[v];               // uniform across block -> 1 cacheline
    float x = *p;                      // coalesced 512B row segment
    p += CIN;
    acc[lab * RED_COLS + tid] += x;    // thread-exclusive LDS column
  }
  __syncthreads();

  float* __restrict__ dst = roi_sum + colbase + tid;
  for (int r = 0; r < NRROI; ++r)
    atomicAdd(dst + r * CIN, acc[r * RED_COLS + tid]);
}

// ---------------------------------------------------------------------------
// 3) xbar[b,i] = mean_t x[b,i,t]   (x viewed as [8,1536,24], t contiguous)
// ---------------------------------------------------------------------------
__global__ __launch_bounds__(256)
void k_xbar(const float* __restrict__ x, float* __restrict__ xbar) {
  int idx = blockIdx.x * 256 + threadIdx.x;
  if (idx >= BB * CIN) return;
  const float* p = x + (long)idx * TT;
  float s = 0.f;
#pragma unroll
  for (int t = 0; t < TT; ++t) s += p[t];
  xbar[idx] = s * (1.0f / TT);
}

// pooled[b,c] = mean over 8 spatial of xbar[b, c*8+d]
__global__ __launch_bounds__(256)
void k_pooled(const float* __restrict__ xbar, float* __restrict__ pooled) {
  int idx = blockIdx.x * 256 + threadIdx.x;
  if (idx >= BB * 192) return;
  int b = idx / 192, c = idx % 192;
  float s = 0.f;
#pragma unroll
  for (int d = 0; d < 8; ++d) s += xbar[b * CIN + c * 8 + d];
  pooled[idx] = s * 0.125f;
}

// ---------------------------------------------------------------------------
// 4) x_node[b,r] = xbar[b,:] . roi_sum[r,:]/cnt[r] + bsum[r]/cnt[r]
// ---------------------------------------------------------------------------
__global__ __launch_bounds__(128)
void k_xnode(const float* __restrict__ xbar, const float* __restrict__ roi_sum,
             const float* __restrict__ cnt, const float* __restrict__ bsum,
             float* __restrict__ x_node) {
  __shared__ float red[BB * 128];
  const int r = blockIdx.x, tid = threadIdx.x;
  float p[BB];
#pragma unroll
  for (int b = 0; b < BB; ++b) p[b] = 0.f;
  for (int i = tid; i < CIN; i += 128) {
    float w = roi_sum[r * CIN + i];
#pragma unroll
    for (int b = 0; b < BB; ++b) p[b] += xbar[b * CIN + i] * w;
  }
#pragma unroll
  for (int b = 0; b < BB; ++b) red[b * 128 + tid] = p[b];
  __syncthreads();
  for (int s = 64; s > 0; s >>= 1) {
    if (tid < s)
#pragma unroll
      for (int b = 0; b < BB; ++b) red[b * 128 + tid] += red[b * 128 + tid + s];
    __syncthreads();
  }
  if (tid < BB) {
    float inv = 1.0f / fmaxf(cnt[r], 1.0f);
    x_node[tid * NRROI + r] = (red[tid * 128] + bsum[r]) * inv;
  }
}

// ---------------------------------------------------------------------------
// 5) 116 tiny per-ROI MLPs: one block per ROI, thread = hidden unit
// ---------------------------------------------------------------------------
__global__ __launch_bounds__(64)
void k_roimlp(const float* __restrict__ x_node,
              const float* __restrict__ w1, const float* __restrict__ b1,
              const float* __restrict__ w2, const float* __restrict__ b2,
              const float* __restrict__ w3, const float* __restrict__ b3,
              float* __restrict__ fx) {
  __shared__ float h1[BB][64];
  __shared__ float red[64][BB];
  const int n = blockIdx.x, o = threadIdx.x;
  float w1o = w1[n * 64 + o], b1o = b1[n * 64 + o];
#pragma unroll
  for (int b = 0; b < BB; ++b)
    h1[b][o] = fmaxf(x_node[b * NRROI + n] * w1o + b1o, 0.f);
  __syncthreads();
  const float* w2row = w2 + ((long)n * 64 + o) * 64;
  float h2[BB];
  float b2o = b2[n * 64 + o];
#pragma unroll
  for (int b = 0; b < BB; ++b) h2[b] = b2o;
  for (int i = 0; i < 64; ++i) {
    float w = w2row[i];
#pragma unroll
    for (int b = 0; b < BB; ++b) h2[b] += h1[b][i] * w;
  }
  float w3o = w3[n * 64 + o];
#pragma unroll
  for (int b = 0; b < BB; ++b) red[o][b] = fmaxf(h2[b], 0.f) * w3o;
  __syncthreads();
  if (o < BB) {
    float s = 0.f;
    for (int i = 0; i < 64; ++i) s += red[i][o];
    fx[o * NRROI + n] = s + b3[n];
  }
}

// ---------------------------------------------------------------------------
// 6) Distance MLP, fully fused, WMMA F32 16x16x4 for the 13456x64x64 GEMM.
//    One wave (32 threads, EXEC all ones) per 16-row tile; 841 tiles exactly.
//    A-layout (ISA 16x4 f32): lane -> m=lane&15, K pair base kb=(lane>>4)*2.
//    B-layout mirrors A with N in place of M (b[j] = m_w2[n][k0+j]).
//    C/D layout: VGPR j, lane -> n=lane&15 (+16*ntile), m = j + 8*(lane>>4).
// ---------------------------------------------------------------------------
__global__ __launch_bounds__(32)
void k_dist_wmma(const float* __restrict__ nd,
                 const float* __restrict__ mw1, const float* __restrict__ mb1,
                 const float* __restrict__ mw2, const float* __restrict__ mb2,
                 const float* __restrict__ mw3, const float* __restrict__ mb3,
                 float* __restrict__ md) {
  const int lane = threadIdx.x;
  const int tile = blockIdx.x;          // 0..840
  const int m  = lane & 15;
  const int kb = (lane >> 4) << 1;      // 0 or 2
  const float dval = nd[tile * 16 + m];

  v8f c0, c1, c2, c3;
  const float bias0 = mb2[ 0 + m], bias1 = mb2[16 + m];
  const float bias2 = mb2[32 + m], bias3 = mb2[48 + m];
#pragma unroll
  for (int j = 0; j < 8; ++j) { c0[j] = bias0; c1[j] = bias1; c2[j] = bias2; c3[j] = bias3; }

#pragma unroll
  for (int ks = 0; ks < 16; ++ks) {
    const int k0 = ks * 4 + kb;
    // layer-1 on the fly, directly in WMMA A layout: g1[m][k]=relu(d*w1+b1)
    v2f a;
    a[0] = fmaxf(dval * mw1[k0]     + mb1[k0],     0.f);
    a[1] = fmaxf(dval * mw1[k0 + 1] + mb1[k0 + 1], 0.f);
    v2f bv;
    bv[0] = mw2[( 0 + m) * 64 + k0]; bv[1] = mw2[( 0 + m) * 64 + k0 + 1];
    c0 = __builtin_amdgcn_wmma_f32_16x16x4_f32(false, a, false, bv, (short)0, c0, false, false);
    bv[0] = mw2[(16 + m) * 64 + k0]; bv[1] = mw2[(16 + m) * 64 + k0 + 1];
    c1 = __builtin_amdgcn_wmma_f32_16x16x4_f32(false, a, false, bv, (short)0, c1, false, false);
    bv[0] = mw2[(32 + m) * 64 + k0]; bv[1] = mw2[(32 + m) * 64 + k0 + 1];
    c2 = __builtin_amdgcn_wmma_f32_16x16x4_f32(false, a, false, bv, (short)0, c2, false, false);
    bv[0] = mw2[(48 + m) * 64 + k0]; bv[1] = mw2[(48 + m) * 64 + k0 + 1];
    c3 = __builtin_amdgcn_wmma_f32_16x16x4_f32(false, a, false, bv, (short)0, c3, false, false);
  }

  // relu(g2) . m_w3 + b3, reduced across the 16 lanes of each half-wave
  const float w3_0 = mw3[m], w3_1 = mw3[16 + m];
  const float w3_2 = mw3[32 + m], w3_3 = mw3[48 + m];
  const float b3v = mb3[0];
#pragma unroll
  for (int j = 0; j < 8; ++j) {
    float p = fmaxf(c0[j], 0.f) * w3_0 + fmaxf(c1[j], 0.f) * w3_1 +
              fmaxf(c2[j], 0.f) * w3_2 + fmaxf(c3[j], 0.f) * w3_3;
    p += __shfl_xor(p, 1);
    p += __shfl_xor(p, 2);
    p += __shfl_xor(p, 4);
    p += __shfl_xor(p, 8);
    if (m == 0) md[tile * 16 + j + ((lane >> 4) << 3)] = p + b3v;
  }
}

// ---------------------------------------------------------------------------
// 7) mmat[j] = fc1_w[j,:] . md + fc1_b[j]   (64 rows of 13456)
// ---------------------------------------------------------------------------
__global__ __launch_bounds__(256)
void k_fc1(const float* __restrict__ fc1_w, const float* __restrict__ fc1_b,
           const float* __restrict__ md, float* __restrict__ mmat) {
  __shared__ float red[256];
  const int j = blockIdx.x, tid = threadIdx.x;
  const float* wrow = fc1_w + (long)j * NPAIR;
  float s = 0.f;
  for (int v = tid; v < NPAIR; v += 256) s += wrow[v] * md[v];
  red[tid] = s;
  __syncthreads();
  for (int k = 128; k > 0; k >>= 1) {
    if (tid < k) red[tid] += red[tid + k];
    __syncthreads();
  }
  if (tid == 0) mmat[j] = red[0] + fc1_b[j];
}

// ---------------------------------------------------------------------------
// 8) mf = mmat(8x8) @ fx(8x116); cat = [mf | pooled]  -> [8,308]
// ---------------------------------------------------------------------------
__global__ __launch_bounds__(128)
void k_mfcat(const float* __restrict__ mmat, const float* __restrict__ fx,
             const float* __restrict__ pooled, float* __restrict__ cat) {
  __shared__ float sm[64];
  __shared__ float sf[BB * NRROI];
  const int tid = threadIdx.x;
  if (tid < 64) sm[tid] = mmat[tid];
  for (int i = tid; i < BB * NRROI; i += 128) sf[i] = fx[i];
  __syncthreads();
  for (int idx = tid; idx < BB * NRROI; idx += 128) {
    int b = idx / NRROI, r = idx % NRROI;
    float s = 0.f;
#pragma unroll
    for (int q = 0; q < 8; ++q) s += sm[b * 8 + q] * sf[q * NRROI + r];
    cat[b * 308 + r] = s;
  }
  for (int idx = tid; idx < BB * 192; idx += 128) {
    int b = idx / 192, c = idx % 192;
    cat[b * 308 + NRROI + c] = pooled[idx];
  }
}

// ---------------------------------------------------------------------------
// 9) hid[b,j] = cat[b,:] . hidden_w[j,:] + hidden_b[j]   (no activation)
// ---------------------------------------------------------------------------
__global__ __launch_bounds__(256)
void k_hidden(const float* __restrict__ cat, const float* __restrict__ hw,
              const float* __restrict__ hb, float* __restrict__ hid) {
  int idx = blockIdx.x * 256 + threadIdx.x;   // 8*768 = 6144
  if (idx >= BB * 768) return;
  int j = idx >> 3, b = idx & 7;
  const float* wrow = hw + (long)j * 308;
  const float* crow = cat + b * 308;
  float s = hb[j];
  for (int k = 0; k < 308; ++k) s += wrow[k] * crow[k];
  hid[b * 768 + j] = s;
}

// ---------------------------------------------------------------------------
// 10) out[b] = hid[b,:] . out_w + out_b
// ---------------------------------------------------------------------------
__global__ __launch_bounds__(256)
void k_out(const float* __restrict__ hid, const float* __restrict__ ow,
           const float* __restrict__ ob, float* __restrict__ out) {
  __shared__ float red[BB * 256];
  const int tid = threadIdx.x;
  float p[BB];
#pragma unroll
  for (int b = 0; b < BB; ++b) p[b] = 0.f;
  for (int j = tid; j < 768; j += 256) {
    float w = ow[j];
#pragma unroll
    for (int b = 0; b < BB; ++b) p[b] += hid[b * 768 + j] * w;
  }
#pragma unroll
  for (int b = 0; b < BB; ++b) red[b * 256 + tid] = p[b];
  __syncthreads();
  for (int k = 128; k > 0; k >>= 1) {
    if (tid < k)
#pragma unroll
      for (int b = 0; b < BB; ++b) red[b * 256 + tid] += red[b * 256 + tid + k];
    __syncthreads();
  }
  if (tid < BB) out[tid] = red[tid * 256] + ob[0];
}

// ---------------------------------------------------------------------------
extern "C" void kernel_launch(void* const* d_in, const int* in_sizes, int n_in,
                              void* d_out, int out_size, void* d_ws, size_t ws_size,
                              hipStream_t stream) {
  (void)in_sizes; (void)n_in; (void)out_size; (void)ws_size;
  const float* x        = (const float*)d_in[0];
  const int*   labels   = (const int*)  d_in[1];
  const float* nd       = (const float*)d_in[2];
  const float* fc_w     = (const float*)d_in[3];
  const float* fc_b     = (const float*)d_in[4];
  const float* fs_w1    = (const float*)d_in[5];
  const float* fs_b1    = (const float*)d_in[6];
  const float* fs_w2    = (const float*)d_in[7];
  const float* fs_b2    = (const float*)d_in[8];
  const float* fs_w3    = (const float*)d_in[9];
  const float* fs_b3    = (const float*)d_in[10];
  const float* m_w1     = (const float*)d_in[11];
  const float* m_b1     = (const float*)d_in[12];
  const float* m_w2     = (const float*)d_in[13];
  const float* m_b2     = (const float*)d_in[14];
  const float* m_w3     = (const float*)d_in[15];
  const float* m_b3     = (const float*)d_in[16];
  const float* fc1_w    = (const float*)d_in[17];
  const float* fc1_b    = (const float*)d_in[18];
  const float* hidden_w = (const float*)d_in[19];
  const float* hidden_b = (const float*)d_in[20];
  const float* out_w    = (const float*)d_in[21];
  const float* out_b    = (const float*)d_in[22];
  float* ws  = (float*)d_ws;
  float* out = (float*)d_out;

  // zero the accumulator region (graph-capture safe)
  hipMemsetAsync(ws, 0, (size_t)ZERO_FLOATS * sizeof(float), stream);

  k_count<<<64, 256, 0, stream>>>(labels, fc_b, ws + OFF_CNT, ws + OFF_BSUM);

  // dominant: 1.67 GB stream of fc_w -> [116,1536] segment sums
  k_reduce_fcw<<<dim3(CIN / RED_COLS, (VOXN + RED_ROWS - 1) / RED_ROWS),
                 RED_COLS, 0, stream>>>(fc_w, labels, ws + OFF_ROI_SUM);

  k_xbar<<<(BB * CIN + 255) / 256, 256, 0, stream>>>(x, ws + OFF_XBAR);
  k_pooled<<<(BB * 192 + 255) / 256, 256, 0, stream>>>(ws + OFF_XBAR, ws + OFF_POOLED);

  k_xnode<<<NRROI, 128, 0, stream>>>(ws + OFF_XBAR, ws + OFF_ROI_SUM,
                                     ws + OFF_CNT, ws + OFF_BSUM, ws + OFF_XNODE);

  k_roimlp<<<NRROI, 64, 0, stream>>>(ws + OFF_XNODE, fs_w1, fs_b1, fs_w2, fs_b2,
                                     fs_w3, fs_b3, ws + OFF_FX);

  k_dist_wmma<<<NPAIR / 16, 32, 0, stream>>>(nd, m_w1, m_b1, m_w2, m_b2,
                                             m_w3, m_b3, ws + OFF_MD);

  k_fc1<<<64, 256, 0, stream>>>(fc1_w, fc1_b, ws + OFF_MD, ws + OFF_MMAT);

  k_mfcat<<<1, 128, 0, stream>>>(ws + OFF_MMAT, ws + OFF_FX, ws + OFF_POOLED,
                                 ws + OFF_CAT);

  k_hidden<<<(BB * 768 + 255) / 256, 256, 0, stream>>>(ws + OFF_CAT, hidden_w,
                                                       hidden_b, ws + OFF_HID);

  k_out<<<1, 256, 0, stream>>>(ws + OFF_HID, out_w, out_b, out);
}